// GNN_63342177681456
// MI455X (gfx1250) — compile-verified
//
#include <hip/hip_runtime.h>
#include <hip/hip_bf16.h>

// ---------------------------------------------------------------------------
// GAT (2-layer, heads=1) + mean-pool + linear head for MI455X (gfx1250).
//
//  - Node GEMMs h = x@W use V_WMMA_F32_16X16X4_F32 (exact f32), one wave per
//    16-node tile. N_NODES % 16 == 0 and inputs are pre-padded to K in {8,16},
//    so the WMMA hot path has ZERO branches / exec-mask toggles.
//  - Attention scalars es = h@a_src fold to es = x@(W a_src).
//  - Per-destination softmax in 2 edge passes (u32 atomic-max with monotone
//    float encoding, then exp/atomic-add accumulation), normalize per node.
//  - Edge indices (25.6MB) + node arrays (~24MB) are L2-resident (192MB).
// ---------------------------------------------------------------------------

typedef __attribute__((ext_vector_type(2))) float v2f;
typedef __attribute__((ext_vector_type(8))) float v8f;

#define N_NODES 100000   // divisible by 16 -> 6250 exact tiles
#define N_GRAPHS 512
#define E_EDGE 3200000
#define E_TOT 3300000
#define NEG_SLOPE 0.2f

// monotone float <-> uint order-preserving encoding (for u32 atomic max)
__device__ __forceinline__ unsigned fenc(float f) {
  unsigned u = __float_as_uint(f);
  return (u & 0x80000000u) ? ~u : (u | 0x80000000u);
}
__device__ __forceinline__ float fdec(unsigned u) {
  return __uint_as_float((u & 0x80000000u) ? (u & 0x7fffffffu) : ~u);
}

// ---- prep: fused attention vectors fv = {Ws1[8], Wd1[8], Ws2[16], Wd2[16]},
//      zero-padded Wpad1[8][16], and zero the pooled region. One block.
__global__ void prep_kernel(const float* __restrict__ W1, const float* __restrict__ as1,
                            const float* __restrict__ ad1, const float* __restrict__ W2,
                            const float* __restrict__ as2, const float* __restrict__ ad2,
                            float* __restrict__ fv, float* __restrict__ Wpad1,
                            float* __restrict__ pooled_region) {
  int t = threadIdx.x;
  if (t < 8) {
    float s1 = 0.f, d1 = 0.f;
    if (t < 5) {
      for (int f = 0; f < 16; ++f) {
        float w = W1[t * 16 + f];
        s1 += w * as1[f];
        d1 += w * ad1[f];
      }
    }
    fv[t] = s1;
    fv[8 + t] = d1;
  }
  if (t < 16) {
    float s2 = 0.f, d2 = 0.f;
    for (int f = 0; f < 16; ++f) {
      float w = W2[t * 16 + f];
      s2 += w * as2[f];
      d2 += w * ad2[f];
    }
    fv[16 + t] = s2;
    fv[32 + t] = d2;
  }
  if (t < 128) Wpad1[t] = (t < 80) ? W1[t] : 0.f;  // [8][16], rows 5..7 zero
  for (int i = t; i < N_GRAPHS * 17; i += blockDim.x) pooled_region[i] = 0.f;
}

// ---- pad x [N,5] -> xpad [N,8] (cols 5..7 zero), once
__global__ void pad_x_kernel(const float* __restrict__ x, float* __restrict__ xpad) {
  const int gid = blockIdx.x * blockDim.x + threadIdx.x;
  if (gid >= N_NODES * 8) return;
  const int i = gid >> 3;
  const int k = gid & 7;
  xpad[gid] = (k < 5) ? x[i * 5 + k] : 0.f;
}

// ---- zero a float region (acc + s + mEnc contiguous)
__global__ void zero_kernel(float* __restrict__ p, int n) {
  int i = blockIdx.x * blockDim.x + threadIdx.x;
  if (i < n) p[i] = 0.f;
}

// ---- node GEMM via WMMA f32 16x16x4 + fused attention scalars.
// One wave per 16-node tile; KD in {8,16}; no guards -> EXEC all-1s throughout.
// A 16x4 layout: lanes 0-15 rows, VGPR pair = (K, K+1), hi half = K+2.
// B 4x16 layout mirrored: row K striped across lanes 0-15, hi half = K+2.
template <int KD>
__global__ void gemm_att_kernel(const float* __restrict__ xin, const float* __restrict__ Wm,
                                const float* __restrict__ wsrc, const float* __restrict__ wdst,
                                float* __restrict__ h, float* __restrict__ es,
                                float* __restrict__ ed) {
  const int lane = threadIdx.x & 31;
  const int wave = threadIdx.x >> 5;
  const int tile = blockIdx.x * (blockDim.x >> 5) + wave;
  const int row0 = tile * 16;
  if (row0 >= N_NODES) return;  // wave-uniform
  const int lr = lane & 15;
  const int hi = lane >> 4;
  const int row = row0 + lr;

  v8f c = {};
#pragma unroll
  for (int ks = 0; ks < KD / 4; ++ks) {
    const int kb = ks * 4 + hi * 2;
    const v2f a = *(const v2f*)(xin + row * KD + kb);  // 8B-aligned (KD,kb even)
    v2f b;
    b.x = Wm[kb * 16 + lr];
    b.y = Wm[(kb + 1) * 16 + lr];
    c = __builtin_amdgcn_wmma_f32_16x16x4_f32(false, a, false, b, (short)0, c, false, false);
  }
#pragma unroll
  for (int r = 0; r < 8; ++r) h[(row0 + hi * 8 + r) * 16 + lr] = c[r];

  // fused attention scalars: es = x . (W a_src), ed = x . (W a_dst)
  if (hi == 0) {
    float a_s = 0.f, a_d = 0.f;
#pragma unroll
    for (int k = 0; k < KD; ++k) {
      const float xv = xin[row * KD + k];
      a_s += xv * wsrc[k];
      a_d += xv * wdst[k];
    }
    es[row] = a_s;
    ed[row] = a_d;
  }
}

// ---- edge pass 1: per-destination max of leaky_relu logits (u32 atomic max)
__global__ void edge_max_kernel(const int* __restrict__ ei, const float* __restrict__ es,
                                const float* __restrict__ ed, unsigned* __restrict__ mEnc) {
  const int gid = blockIdx.x * blockDim.x + threadIdx.x;
  if (gid >= E_TOT) return;
  int src, dst;
  if (gid < E_EDGE) {
    src = ei[gid];
    dst = ei[E_EDGE + gid];
  } else {
    src = dst = gid - E_EDGE;
  }
  float v = es[src] + ed[dst];
  v = (v > 0.f) ? v : NEG_SLOPE * v;
  atomicMax(&mEnc[dst], fenc(v));
}

// ---- edge pass 2: ex = exp(e - m[dst]); s[dst] += ex; acc[dst][f] += ex*h[src][f]
// 16 threads per edge (lane f handles feature f) -> coalesced h[src] gather.
__global__ void edge_acc_kernel(const int* __restrict__ ei, const float* __restrict__ es,
                                const float* __restrict__ ed, const unsigned* __restrict__ mEnc,
                                const float* __restrict__ h, float* __restrict__ s,
                                float* __restrict__ acc) {
  const unsigned gid = blockIdx.x * blockDim.x + threadIdx.x;
  const unsigned e = gid >> 4;
  const int f = gid & 15;
  if (e >= E_TOT) return;
  int src, dst;
  if (e < E_EDGE) {
    src = ei[e];
    dst = ei[E_EDGE + e];
  } else {
    src = dst = (int)e - E_EDGE;
  }
  float v = es[src] + ed[dst];
  v = (v > 0.f) ? v : NEG_SLOPE * v;
  const float ex = __expf(v - fdec(mEnc[dst]));
  if (f == 0) atomicAdd(&s[dst], ex);
  atomicAdd(&acc[dst * 16 + f], ex * h[src * 16 + f]);
}

// ---- per-node finalize: xb = relu(acc/s + b), layer2 also pools into pooled/cnt
//      layer-2 input is consumed at stride 16 so no extra padding needed.
__global__ void node_fin_kernel(const float* __restrict__ acc, const float* __restrict__ s,
                                const float* __restrict__ bias, float* __restrict__ xb,
                                const int* __restrict__ batch, float* __restrict__ pooled,
                                float* __restrict__ cnt, int doPool) {
  const int gid = blockIdx.x * blockDim.x + threadIdx.x;
  if (gid >= N_NODES * 16) return;
  const int i = gid >> 4;
  const int f = gid & 15;
  float v = acc[gid] / (s[i] + 1e-16f) + bias[f];
  v = fmaxf(v, 0.f);
  xb[gid] = v;
  if (doPool) {
    const int g = batch[i];
    atomicAdd(&pooled[g * 16 + f], v);
    if (f == 0) atomicAdd(&cnt[g], 1.0f);
  }
}

// ---- head: out[g][o] = (pooled[g]/max(cnt,1)) @ Wf + bf
__global__ void head_kernel(const float* __restrict__ pooled, const float* __restrict__ cnt,
                            const float* __restrict__ Wf, const float* __restrict__ bf,
                            float* __restrict__ out) {
  const int gid = blockIdx.x * blockDim.x + threadIdx.x;
  if (gid >= N_GRAPHS * 2) return;
  const int g = gid >> 1;
  const int o = gid & 1;
  const float c = fmaxf(cnt[g], 1.0f);
  float a = 0.f;
#pragma unroll
  for (int f = 0; f < 16; ++f) a += (pooled[g * 16 + f] / c) * Wf[f * 2 + o];
  out[gid] = a + bf[o];
}

extern "C" void kernel_launch(void* const* d_in, const int* in_sizes, int n_in,
                              void* d_out, int out_size, void* d_ws, size_t ws_size,
                              hipStream_t stream) {
  const float* x = (const float*)d_in[0];
  const int* ei = (const int*)d_in[1];
  const int* batch = (const int*)d_in[2];
  const float* W1 = (const float*)d_in[3];
  const float* as1 = (const float*)d_in[4];
  const float* ad1 = (const float*)d_in[5];
  const float* b1 = (const float*)d_in[6];
  const float* W2 = (const float*)d_in[7];
  const float* as2 = (const float*)d_in[8];
  const float* ad2 = (const float*)d_in[9];
  const float* b2 = (const float*)d_in[10];
  const float* Wf = (const float*)d_in[11];
  const float* bf = (const float*)d_in[12];
  float* out = (float*)d_out;

  // workspace layout (floats)
  float* ws = (float*)d_ws;
  float* h = ws;                              // N*16
  float* xb = h + N_NODES * 16;               // N*16
  float* acc = xb + N_NODES * 16;             // N*16   } contiguous zero region
  float* s = acc + N_NODES * 16;              // N      }
  unsigned* mEnc = (unsigned*)(s + N_NODES);  // N      }
  float* es = (float*)(mEnc + N_NODES);       // N
  float* ed = es + N_NODES;                   // N
  float* xpad = ed + N_NODES;                 // N*8
  float* pooled = xpad + N_NODES * 8;         // G*16
  float* cnt = pooled + N_GRAPHS * 16;        // G
  float* fv = cnt + N_GRAPHS;                 // 48
  float* Wpad1 = fv + 48;                     // 128

  const int ZERO_N = N_NODES * 18;  // acc + s + mEnc
  const int TILES = N_NODES / 16;   // exact
  const dim3 blk(256);
  const dim3 gGemm((TILES + 7) / 8);
  const dim3 gZero((ZERO_N + 255) / 256);
  const dim3 gEdge((E_TOT + 255) / 256);
  const dim3 gEdge16((E_TOT * 16 + 255) / 256);
  const dim3 gNode((N_NODES * 16 + 255) / 256);

  prep_kernel<<<1, blk, 0, stream>>>(W1, as1, ad1, W2, as2, ad2, fv, Wpad1, pooled);
  pad_x_kernel<<<(N_NODES * 8 + 255) / 256, blk, 0, stream>>>(x, xpad);

  // ---- layer 1 (K padded 5->8)
  zero_kernel<<<gZero, blk, 0, stream>>>(acc, ZERO_N);
  gemm_att_kernel<8><<<gGemm, blk, 0, stream>>>(xpad, Wpad1, fv, fv + 8, h, es, ed);
  edge_max_kernel<<<gEdge, blk, 0, stream>>>(ei, es, ed, mEnc);
  edge_acc_kernel<<<gEdge16, blk, 0, stream>>>(ei, es, ed, mEnc, h, s, acc);
  node_fin_kernel<<<gNode, blk, 0, stream>>>(acc, s, b1, xb, batch, pooled, cnt, 0);

  // ---- layer 2 (K = 16)
  zero_kernel<<<gZero, blk, 0, stream>>>(acc, ZERO_N);
  gemm_att_kernel<16><<<gGemm, blk, 0, stream>>>(xb, W2, fv + 16, fv + 32, h, es, ed);
  edge_max_kernel<<<gEdge, blk, 0, stream>>>(ei, es, ed, mEnc);
  edge_acc_kernel<<<gEdge16, blk, 0, stream>>>(ei, es, ed, mEnc, h, s, acc);
  node_fin_kernel<<<gNode, blk, 0, stream>>>(acc, s, b2, xb, batch, pooled, cnt, 1);

  // ---- mean pool + linear head
  head_kernel<<<(N_GRAPHS * 2 + 255) / 256, blk, 0, stream>>>(pooled, cnt, Wf, bf, out);

  (void)in_sizes; (void)n_in; (void)out_size; (void)ws_size;
}